// DFHLoss_26474178412717
// MI455X (gfx1250) — compile-verified
//
#include <hip/hip_runtime.h>

#define BIT     64
#define NCLS    100
#define NPAD    112          // logical padded class dim (7 x 16 tiles)
#define SV      116          // LDS storage stride for V (bank-conflict-free)
#define BSTRIDE 144          // LDS storage stride for metric B tile
#define NTRAIN  100000
#define BATCH   256

// ws layout (bytes)
#define OFF_LABELS 0                       // int[256]
#define OFF_INV    1024                    // int[100000]
#define OFF_LABT   401024                  // int[100000]
#define OFF_B      801024                  // float[64*256]
#define OFF_S      866560                  // float[64*100]
#define OFF_N      892160                  // float[112]
#define OFF_PART   892608                  // float[<=800]
#define OFF_GM     896000                  // float[112*112]

typedef float v2f __attribute__((ext_vector_type(2)));
typedef float v8f __attribute__((ext_vector_type(8)));

static __device__ __forceinline__ v8f wmma_f32_4(v2f a, v2f b, v8f c) {
  // D = A(16x4) * B(4x16) + C, f32, wave32
  return __builtin_amdgcn_wmma_f32_16x16x4_f32(false, a, false, b, (short)0, c,
                                               false, false);
}

static __device__ __forceinline__ float sgnf(float x) {
  return (x > 0.f) ? 1.f : ((x < 0.f) ? -1.f : 0.f);
}
static __device__ __forceinline__ float softplusf(float x) {
  return (x > 20.f) ? x : log1pf(__expf(x));   // log1p(exp(x)), x <= 101 by clip
}
static __device__ __forceinline__ unsigned lds_off(const void* p) {
  // generic LDS pointer: low 32 bits are the wave-relative LDS byte offset
  return (unsigned)(unsigned long long)p;
}
static __device__ __forceinline__ void async_ld_b32(unsigned dst_lds, const void* gaddr) {
  asm volatile("global_load_async_to_lds_b32 %0, %1, off"
               :: "v"(dst_lds), "v"(gaddr) : "memory");
}
static __device__ __forceinline__ void wait_async0() {
  asm volatile("s_wait_asynccnt 0x0" ::: "memory");
}

// ---------- tiny setup kernels ----------
__global__ void k_init_inv(int* __restrict__ inv) {
  int t = blockIdx.x * 256 + threadIdx.x;
  if (t < NTRAIN) inv[t] = -1;
}

__global__ void k_labels(const float* __restrict__ y, const int* __restrict__ ind,
                         int* __restrict__ labels, int* __restrict__ inv) {
  int j = threadIdx.x;                       // 256 threads
  const float* row = y + j * NCLS;
  int best = 0; float bv = row[0];
  for (int c = 1; c < NCLS; ++c) { float v = row[c]; if (v > bv) { bv = v; best = c; } }
  labels[j] = best;
  inv[ind[j]] = j;
}

// b[i,j] = sign(sign(V[i,lab_j]) + u[j,i]);  S[i,c] = sum_{j:lab_j=c} b[i,j];  n_c
__global__ void k_b_s(const float* __restrict__ u, const float* __restrict__ V,
                      const int* __restrict__ labels, float* __restrict__ b_ws,
                      float* __restrict__ S_ws, float* __restrict__ n_ws) {
  __shared__ float brow[BATCH];
  __shared__ int   labs[BATCH];
  int i = blockIdx.x;                        // 0..63
  int j = threadIdx.x;                       // 0..255
  int lab = labels[j];
  labs[j] = lab;
  float bij = sgnf(1.0f /*MU*/ * sgnf(V[i * NCLS + lab]) + u[j * BIT + i]);
  b_ws[i * BATCH + j] = bij;
  brow[j] = bij;
  __syncthreads();
  if (j < NCLS) {
    float s = 0.f;
    for (int jj = 0; jj < BATCH; ++jj) if (labs[jj] == j) s += brow[jj];
    S_ws[i * NCLS + j] = s;
    if (i == 0) {
      float n = 0.f;
      for (int jj = 0; jj < BATCH; ++jj) if (labs[jj] == j) n += 1.f;
      n_ws[j] = n;
    }
  }
}

// per-train-column label (argmax of scattered Y)
__global__ void k_labT(const float* __restrict__ Y, const int* __restrict__ inv,
                       const int* __restrict__ labels, int* __restrict__ labT) {
  int t = blockIdx.x * 256 + threadIdx.x;
  if (t >= NTRAIN) return;
  int iv = inv[t];
  int lab;
  if (iv >= 0) {
    lab = labels[iv];
  } else {
    int best = 0; float bv = Y[t];
    for (int c = 1; c < NCLS; ++c) {
      float v = Y[(size_t)c * NTRAIN + t];
      if (v > bv) { bv = v; best = c; }
    }
    lab = best;
  }
  labT[t] = lab;
}

// ---------- 200-step codebook SGD, single workgroup, f32 WMMA ----------
__global__ void __launch_bounds__(1024)
k_center(const float* __restrict__ Vg, const float* __restrict__ TKg,
         const float* __restrict__ S_ws, const float* __restrict__ n_ws,
         float* __restrict__ Gmg, float* __restrict__ outV) {
  __shared__ float Va[BIT * SV];
  __shared__ float Vb[BIT * SV];
  const int tid = threadIdx.x;
  for (int e = tid; e < BIT * SV; e += 1024) {
    int i = e / SV, c = e - i * SV;
    Va[e] = (c < NCLS) ? Vg[i * NCLS + c] : 0.f;
  }
  __syncthreads();

  const int wave = tid >> 5;
  const int lane = tid & 31;
  const int lh = lane & 15;
  const int lg = lane >> 4;                  // 0 or 1
  const float cIntra = 2.f / (float)(BIT * BATCH);
  const float cInter = 4.f / (float)(NCLS * NCLS);  // VUL = 1
  const float cQuant = 2.f / (float)(BIT * NCLS);   // NTA = 1

  float* cur = Va;
  float* nxt = Vb;
  for (int it = 0; it < 200; ++it) {
    float alpha = 0.03f;
    if (it >= 149) alpha *= 0.1f;
    if (it >= 179) alpha *= 0.1f;

    // phase 1: Gm = V^T V - TK (padded 112x112 -> global ws), 7x7 tiles / 32 waves
    for (int tile = wave; tile < 49; tile += 32) {
      int tm = tile / 7, tn = tile - tm * 7;
      v8f acc = {0.f, 0.f, 0.f, 0.f, 0.f, 0.f, 0.f, 0.f};
      int am = tm * 16 + lh;
      int bn = tn * 16 + lh;
      for (int k0 = 0; k0 < BIT; k0 += 4) {
        v2f a, b;
        int ak = k0 + (lg << 1);
        a.x = cur[ak * SV + am];
        a.y = cur[(ak + 1) * SV + am];
        int bk = k0 + lg;
        b.x = cur[bk * SV + bn];
        b.y = cur[(bk + 2) * SV + bn];
        acc = wmma_f32_4(a, b, acc);
      }
#pragma unroll
      for (int r = 0; r < 8; ++r) {
        int gm = tm * 16 + r + (lg << 3);
        int gn = tn * 16 + lh;
        float tk = (gm < NCLS && gn < NCLS) ? TKg[gm * NCLS + gn] : 0.f;
        Gmg[gm * NPAD + gn] = acc[r] - tk;
      }
    }
    __syncthreads();

    // phase 2 (fused update): Gn = V @ Gm ; V' = V - alpha*grad  (4x7 tiles)
    if (wave < 28) {
      int tm = wave / 7, tn = wave - tm * 7;
      v8f acc = {0.f, 0.f, 0.f, 0.f, 0.f, 0.f, 0.f, 0.f};
      int am = tm * 16 + lh;                 // bit index (<64)
      int bn = tn * 16 + lh;                 // class index
      for (int k0 = 0; k0 < NPAD; k0 += 4) {
        v2f a, b;
        int ak = k0 + (lg << 1);
        a.x = cur[am * SV + ak];
        a.y = cur[am * SV + ak + 1];
        int bk = k0 + lg;
        b.x = Gmg[bk * NPAD + bn];
        b.y = Gmg[(bk + 2) * NPAD + bn];
        acc = wmma_f32_4(a, b, acc);
      }
#pragma unroll
      for (int r = 0; r < 8; ++r) {
        int gm = tm * 16 + r + (lg << 3);    // bit i
        int gn = tn * 16 + lh;               // class c
        float v = cur[gm * SV + gn];
        float nv = 0.f;
        if (gn < NCLS) {
          float g = cIntra * (n_ws[gn] * v - S_ws[gm * NCLS + gn])
                  + cInter * acc[r]
                  + cQuant * (v - sgnf(v));
          nv = v - alpha * g;
        }
        nxt[gm * SV + gn] = nv;
      }
    }
    __syncthreads();
    float* t = cur; cur = nxt; nxt = t;
  }

  for (int e = tid; e < BIT * NCLS; e += 1024) {
    int i = e / NCLS, c = e - i * NCLS;
    outV[e] = cur[i * SV + c];
  }
}

// ---------- metric loss: ip = 0.5*(u @ U_scattered), WMMA f32, async LDS stage ----------
// Each wave owns two 16-row tiles; one B-fragment load feeds two WMMAs.
__global__ void __launch_bounds__(256)
k_metric(const float* __restrict__ u, const float* __restrict__ U,
         const int* __restrict__ ind, const int* __restrict__ labu_g,
         const int* __restrict__ labT_g, float* __restrict__ partials) {
  __shared__ float Bt[BIT * BSTRIDE];        // 64 x 128 tile, stride 144
  __shared__ int   labuL[BATCH];
  __shared__ int   labTL[128];
  __shared__ float red[256];
  const int tid = threadIdx.x, bid = blockIdx.x;
  const int tbase = bid * 128;

  // stage B tile from U via CDNA5 async global->LDS copies
  for (int e = tid; e < BIT * 128; e += 256) {
    int k = e >> 7, tl = e & 127;
    int t = tbase + tl;
    if (t < NTRAIN)
      async_ld_b32(lds_off(&Bt[k * BSTRIDE + tl]), &U[(size_t)k * NTRAIN + t]);
    else
      Bt[k * BSTRIDE + tl] = 0.f;
  }
  labuL[tid] = labu_g[tid];
  if (tid < 128) {
    int t = tbase + tid;
    labTL[tid] = (t < NTRAIN) ? labT_g[t] : -1234567;
  }
  if (tid < BIT) {                           // prefetch next slab
    int tn = (tbase + 128 < NTRAIN) ? (tbase + 128) : 0;
    __builtin_prefetch(&U[(size_t)tid * NTRAIN + tn], 0, 1);
  }
  wait_async0();
  __syncthreads();
  // patch scattered columns (U[:,ind[k]] := u[k,:])
  {
    int t = ind[tid];
    if (t >= tbase && t < tbase + 128) {
      int tl = t - tbase;
      for (int k = 0; k < BIT; ++k) Bt[k * BSTRIDE + tl] = u[tid * BIT + k];
    }
  }
  __syncthreads();

  const int wave = tid >> 5, lane = tid & 31, lh = lane & 15, lg = lane >> 4;
  float lsum = 0.f;

  // A fragments for both row-tiles of this wave (global u is L2-resident)
  const int am0 = wave * 16 + lh;            // rows 0..127
  const int am1 = (wave + 8) * 16 + lh;      // rows 128..255
  v2f afr0[16], afr1[16];
#pragma unroll
  for (int ks = 0; ks < 16; ++ks) {
    int ak = ks * 4 + (lg << 1);
    afr0[ks].x = u[am0 * BIT + ak];
    afr0[ks].y = u[am0 * BIT + ak + 1];
    afr1[ks].x = u[am1 * BIT + ak];
    afr1[ks].y = u[am1 * BIT + ak + 1];
  }

  for (int nt = 0; nt < 8; ++nt) {
    v8f acc0 = {0.f, 0.f, 0.f, 0.f, 0.f, 0.f, 0.f, 0.f};
    v8f acc1 = {0.f, 0.f, 0.f, 0.f, 0.f, 0.f, 0.f, 0.f};
    int bn = nt * 16 + lh;
#pragma unroll
    for (int ks = 0; ks < 16; ++ks) {
      v2f bf;
      int bk = ks * 4 + lg;
      bf.x = Bt[bk * BSTRIDE + bn];
      bf.y = Bt[(bk + 2) * BSTRIDE + bn];
      acc0 = wmma_f32_4(afr0[ks], bf, acc0); // shared B fragment
      acc1 = wmma_f32_4(afr1[ks], bf, acc1);
    }
    int nloc = nt * 16 + lh;
    int tg = tbase + nloc;
    if (tg < NTRAIN) {
      int lt = labTL[nloc];
#pragma unroll
      for (int r = 0; r < 8; ++r) {
        int m0 = wave * 16 + r + (lg << 3);
        float ip0 = fminf(fmaxf(0.5f * acc0[r], -100.f), 50.f);
        lsum += (labuL[m0] == lt) ? softplusf(1.f - ip0) : softplusf(1.f + ip0);
        int m1 = (wave + 8) * 16 + r + (lg << 3);
        float ip1 = fminf(fmaxf(0.5f * acc1[r], -100.f), 50.f);
        lsum += (labuL[m1] == lt) ? softplusf(1.f - ip1) : softplusf(1.f + ip1);
      }
    }
  }
  red[tid] = lsum;
  __syncthreads();
  for (int s = 128; s > 0; s >>= 1) {
    if (tid < s) red[tid] += red[tid + s];
    __syncthreads();
  }
  if (tid == 0) partials[bid] = red[0];
}

// ---------- final reduction: loss = metric/(B*NT) + 0.5 * quant/(64*256) ----------
__global__ void __launch_bounds__(256)
k_final(const float* __restrict__ partials, int nblk, const float* __restrict__ b_ws,
        const float* __restrict__ u, float* __restrict__ out) {
  __shared__ float r1[256], r2[256];
  int tid = threadIdx.x;
  float ms = 0.f;
  for (int i = tid; i < nblk; i += 256) ms += partials[i];
  float qs = 0.f;
  for (int e = tid; e < BIT * BATCH; e += 256) {
    int i = e >> 8, j = e & 255;
    float d = b_ws[e] - u[j * BIT + i];
    qs += d * d;
  }
  r1[tid] = ms; r2[tid] = qs;
  __syncthreads();
  for (int s = 128; s > 0; s >>= 1) {
    if (tid < s) { r1[tid] += r1[tid + s]; r2[tid] += r2[tid + s]; }
    __syncthreads();
  }
  if (tid == 0)
    out[0] = r1[0] * (1.0f / ((float)BATCH * (float)NTRAIN))
           + 0.5f * (r2[0] * (1.0f / (float)(BIT * BATCH)));
}

extern "C" void kernel_launch(void* const* d_in, const int* in_sizes, int n_in,
                              void* d_out, int out_size, void* d_ws, size_t ws_size,
                              hipStream_t stream) {
  const float* u   = (const float*)d_in[0];
  const float* y   = (const float*)d_in[1];
  const int*   ind = (const int*)d_in[2];
  const float* U   = (const float*)d_in[3];
  const float* Y   = (const float*)d_in[4];
  const float* V   = (const float*)d_in[5];
  const float* TK  = (const float*)d_in[6];
  float* out = (float*)d_out;

  char* ws = (char*)d_ws;
  int*   labels = (int*)(ws + OFF_LABELS);
  int*   inv    = (int*)(ws + OFF_INV);
  int*   labT   = (int*)(ws + OFF_LABT);
  float* b_ws   = (float*)(ws + OFF_B);
  float* S_ws   = (float*)(ws + OFF_S);
  float* n_ws   = (float*)(ws + OFF_N);
  float* parts  = (float*)(ws + OFF_PART);
  float* Gmg    = (float*)(ws + OFF_GM);

  (void)in_sizes; (void)n_in; (void)out_size; (void)ws_size;

  k_init_inv<<<(NTRAIN + 255) / 256, 256, 0, stream>>>(inv);
  k_labels<<<1, 256, 0, stream>>>(y, ind, labels, inv);
  k_b_s<<<BIT, 256, 0, stream>>>(u, V, labels, b_ws, S_ws, n_ws);
  k_labT<<<(NTRAIN + 255) / 256, 256, 0, stream>>>(Y, inv, labels, labT);
  k_center<<<1, 1024, 0, stream>>>(V, TK, S_ws, n_ws, Gmg, out + 1);
  int nblkM = (NTRAIN + 127) / 128;
  k_metric<<<nblkM, 256, 0, stream>>>(u, U, ind, labels, labT, parts);
  k_final<<<1, 256, 0, stream>>>(parts, nblkM, b_ws, u, out);
}